// LaplacianLoss_49684181680736
// MI455X (gfx1250) — compile-verified
//
#include <hip/hip_runtime.h>
#include <hip/hip_bf16.h>

// ---------------------------------------------------------------------------
// LaplacianLoss: 7-point stencil residual + global mean-of-squares -> scalar.
// Memory-bound (≈130 MB compulsory traffic -> ~5.6 us floor at 23.3 TB/s).
// Stage 1: z-marching stencil, rolling registers for z-neighbors, LDS-staged
//          center row for x-neighbors, per-block partial sums (deterministic).
// Stage 2: single-block reduction of partials -> d_out[0].
// Wave-level reductions use V_WMMA_F32_16X16X4_F32 with an all-ones A matrix:
// D[m][n] = sum_k B[k][n], so summing D row 0 over n gives the sum of all 32
// lane partials regardless of the B slot<->element bijection.
// ---------------------------------------------------------------------------

#define NN   128
#define NN2  (128 * 128)
#define NN3  (128 * 128 * 128)
#define ZCHUNK 32
#define NZCHUNKS 4
#define INV_H2 16384.0f            // 1/dx^2 = 128^2
#define NPART (NZCHUNKS * 126 * 8) // 4032 stage-1 partials
#define COUNT 16003008.0f          // 8 * 126^3 interior points
#define SCALE (0.2f / COUNT)       // LAPL_WEIGHT * LX^2LY^2LZ^2 / count

typedef __attribute__((ext_vector_type(2))) float v2f;
typedef __attribute__((ext_vector_type(8))) float v8f;

// Sum of `v` across all 32 lanes of the wave, via WMMA f32 16x16x4.
// Requires EXEC == all ones (call from non-divergent code only).
__device__ inline float wave_reduce_wmma(float v) {
    v2f a; a[0] = 1.0f; a[1] = 1.0f;   // A = ones(16x4), layout-independent
    v2f b; b[0] = v;    b[1] = 0.0f;   // each lane partial occupies one B slot
    v8f c = {};
    // 8 args: (neg_a, A, neg_b, B, c_mod, C, reuse_a, reuse_b)
    v8f d = __builtin_amdgcn_wmma_f32_16x16x4_f32(
        false, a, false, b, (short)0, c, false, false);
    // d[0] = column-sum for n = lane%16 (rows M=0 / M=8 of D, identical).
    float col = d[0];
    col += __shfl_xor(col, 1);
    col += __shfl_xor(col, 2);
    col += __shfl_xor(col, 4);
    col += __shfl_xor(col, 8);
    return col;  // total over the 32-lane wave, in every lane
}

__global__ void lapl_stage1(const float* __restrict__ f,
                            const float* __restrict__ g,
                            float* __restrict__ partials) {
    __shared__ float srow[NN];   // center row (z, y) staged for x-neighbors
    __shared__ float swave[4];   // per-wave totals

    const int x  = threadIdx.x;          // 0..127
    const int y  = blockIdx.y + 1;       // 1..126
    const int b  = blockIdx.z;           // 0..7
    const int z0 = 1 + (int)blockIdx.x * ZCHUNK;
    const int z1 = (z0 + ZCHUNK - 1 < 126) ? (z0 + ZCHUNK - 1) : 126;

    const size_t base = (size_t)b * NN3 + (size_t)y * NN + (size_t)x; // (b,0,y,x)

    float fm = f[base + (size_t)(z0 - 1) * NN2];  // plane z-1
    float fc = f[base + (size_t)z0 * NN2];        // plane z
    srow[x] = fc;
    __syncthreads();

    const bool active = (x >= 1) && (x <= 126);
    float acc = 0.0f;

    for (int z = z0; z <= z1; ++z) {
        const size_t zoff = base + (size_t)z * NN2;
        float fp  = f[zoff + NN2];   // plane z+1 (becomes next center)
        float fyl = f[zoff - NN];    // y-1 row (L2/WGP$ hit)
        float fyr = f[zoff + NN];    // y+1 row
        float gg  = g[zoff];         // data, read exactly once

        if (active) {
            float xl  = srow[x - 1];
            float xr  = srow[x + 1];
            float lap = INV_H2 * (fm + fp + fyl + fyr + xl + xr - 6.0f * fc);
            float r   = lap + gg;    // lap - (-data)
            acc = fmaf(r, r, acc);
        }
        __syncthreads();             // all reads of srow done
        srow[x] = fp;                // stage next center row
        __syncthreads();
        fm = fc;
        fc = fp;
    }

    // Non-divergent from here: all 128 threads participate (EXEC all ones).
    float wsum = wave_reduce_wmma(acc);
    const int lane = x & 31, wave = x >> 5;
    if (lane == 0) swave[wave] = wsum;
    __syncthreads();
    if (x == 0) {
        float t = swave[0] + swave[1] + swave[2] + swave[3];
        int pidx = ((int)blockIdx.z * (int)gridDim.y + (int)blockIdx.y)
                       * (int)gridDim.x + (int)blockIdx.x;
        partials[pidx] = t;
    }
}

__global__ void lapl_stage2(const float* __restrict__ partials, int n,
                            float* __restrict__ out) {
    __shared__ float swave[32];
    float acc = 0.0f;
    for (int i = threadIdx.x; i < n; i += blockDim.x)
        acc += partials[i];

    float wsum = wave_reduce_wmma(acc);
    const int lane = threadIdx.x & 31, wave = threadIdx.x >> 5;
    if (lane == 0) swave[wave] = wsum;
    __syncthreads();
    if (threadIdx.x == 0) {
        float t = 0.0f;
        const int nw = blockDim.x >> 5;
        for (int i = 0; i < nw; ++i) t += swave[i];
        out[0] = t * SCALE;
    }
}

extern "C" void kernel_launch(void* const* d_in, const int* in_sizes, int n_in,
                              void* d_out, int out_size, void* d_ws, size_t ws_size,
                              hipStream_t stream) {
    (void)in_sizes; (void)n_in; (void)out_size; (void)ws_size;
    const float* f = (const float*)d_in[0];   // 'output' (8,1,128,128,128) f32
    const float* g = (const float*)d_in[1];   // 'data'   (8,1,128,128,128) f32
    float* partials = (float*)d_ws;           // NPART floats of scratch

    dim3 grid(NZCHUNKS, 126, 8);
    dim3 block(128, 1, 1);
    lapl_stage1<<<grid, block, 0, stream>>>(f, g, partials);
    lapl_stage2<<<1, 1024, 0, stream>>>(partials, NPART, (float*)d_out);
}